// PointLoss_79800492360091
// MI455X (gfx1250) — compile-verified
//
#include <hip/hip_runtime.h>

typedef float v2f __attribute__((ext_vector_type(2)));
typedef float v8f __attribute__((ext_vector_type(8)));

#define BB 8
#define NN 32768
#define KK 64
#define SOFT_SCALE 100.0f
#define LEVEL_SET (-0.07f)
#define LOG2E 1.4426950408889634f

// ---------------------------------------------------------------------------
// Kernel 1: build per-Gaussian quadratic-form coefficients Qs (16 x K per batch)
//   dist2_{n,k} = sum_{u,v} phat_u phat_v Q_k[u,v],  phat=[x,y,z,1]
//   Qs = Q * (-0.5 * log2 e)  so  exp(-0.5 dist2) = exp2(f . Qs_vec)
// Layout: coef[(b*16 + (u*4+v))*KK + k]  (row-major 16 x 64 per batch)
// ---------------------------------------------------------------------------
__global__ void setup_coef_kernel(const float* __restrict__ scales,
                                  const float* __restrict__ rotations,
                                  const float* __restrict__ centers,
                                  float* __restrict__ coef) {
  int idx = blockIdx.x * blockDim.x + threadIdx.x;
  if (idx >= BB * KK) return;
  int b = idx / KK, k = idx % KK;
  const float* rpy = rotations + (b * KK + k) * 3;
  const float* sc  = scales    + (b * KK + k) * 3;
  const float* ct  = centers   + (b * KK + k) * 3;
  float cx = cosf(rpy[0]), cy = cosf(rpy[1]), cz = cosf(rpy[2]);
  float sx = sinf(rpy[0]), sy = sinf(rpy[1]), sz = sinf(rpy[2]);
  float R[3][3];
  R[0][0] = cz * cy; R[0][1] = cz * sy * sx - sz * cx; R[0][2] = cz * sy * cx + sz * sx;
  R[1][0] = sz * cy; R[1][1] = sz * sy * sx + cz * cx; R[1][2] = sz * sy * cx - cz * sx;
  R[2][0] = -sy;     R[2][1] = cy * sx;                R[2][2] = cy * cx;
  // local_i = (sum_j R[j][i] * (p_j - c_j)) / s_i   ->  A (3x4), row i = [M_i | t_i]
  float A[3][4];
  #pragma unroll
  for (int i = 0; i < 3; ++i) {
    float inv = 1.0f / sc[i];
    float m0 = R[0][i] * inv, m1 = R[1][i] * inv, m2 = R[2][i] * inv;
    A[i][0] = m0; A[i][1] = m1; A[i][2] = m2;
    A[i][3] = -(m0 * ct[0] + m1 * ct[1] + m2 * ct[2]);
  }
  const float qs = -0.5f * LOG2E;
  #pragma unroll
  for (int u = 0; u < 4; ++u)
    #pragma unroll
    for (int v = 0; v < 4; ++v) {
      float q = A[0][u] * A[0][v] + A[1][u] * A[1][v] + A[2][u] * A[2][v];
      coef[(b * 16 + (u * 4 + v)) * KK + k] = q * qs;
    }
}

// ---------------------------------------------------------------------------
// Kernel 2: main. One wave handles 16-point tiles; per tile:
//   A-frag = quadratic features of 16 points (16x16 feature matrix, K-chunks of 4)
//   B-frag = coefficient matrix 16x64 (4 N-tiles of 16 Gaussians, hoisted)
//   D = f32 WMMA accumulation over 4 chained 16x16x4 ops -> scaled dist2
//   epilogue: const*exp2 (raw v_exp_f32), butterfly transpose-reduce over k,
//             one sigmoid per lane, masked weighted sq-error accumulate.
// Per-wave partial sums -> workspace (deterministic final reduction).
// ---------------------------------------------------------------------------
__global__ __launch_bounds__(256)
void rbf_loss_kernel(const float* __restrict__ pts,     // B*N*4
                     const float* __restrict__ consts,  // B*K
                     const float* __restrict__ coef,    // B*16*K
                     float* __restrict__ partials) {    // B * wavesPerBatch
  const int b = blockIdx.x;
  const int lane = threadIdx.x & 31;
  const int wavesPerBlock = blockDim.x >> 5;
  const int waveId = blockIdx.y * wavesPerBlock + (threadIdx.x >> 5);
  const int wavesPerBatch = gridDim.y * wavesPerBlock;
  const int nTiles = NN / 16;
  const bool hi = lane >= 16;
  const int l15 = lane & 15;

  // Hoist B fragments: 4 N-tiles (16 Gaussians each) x 4 K-chunks (4 feature rows)
  // B layout (4x16, 32-bit): VGPR v, lanes 0-15 -> row v, lanes 16-31 -> row v+2
  v2f bfrag[4][4];
  const float* cb = coef + b * 16 * KK;
  #pragma unroll
  for (int t = 0; t < 4; ++t)
    #pragma unroll
    for (int q = 0; q < 4; ++q) {
      int row0 = 4 * q + (hi ? 2 : 0);
      int col = t * 16 + l15;
      bfrag[t][q].x = cb[(row0 + 0) * KK + col];
      bfrag[t][q].y = cb[(row0 + 1) * KK + col];
    }
  float cst[4];
  #pragma unroll
  for (int t = 0; t < 4; ++t) cst[t] = consts[b * KK + t * 16 + l15];

  float loss_acc = 0.0f;

  for (int tile = waveId; tile < nTiles; tile += wavesPerBatch) {
    const float4 p = ((const float4*)pts)[(long)b * NN + tile * 16 + l15];
    // A-frag (16x16 features, 32-bit A layout): VGPR v, lane<16 -> K=4q+v,
    // lane>=16 -> K=4q+v+2; feature f_K = phat[K/4] * phat[K%4], phat=[x,y,z,1]
    float pl0 = hi ? p.z : p.x;   // phat[hi?2:0]
    float pl1 = hi ? 1.0f : p.y;  // phat[hi?3:1]
    v2f afrag[4];
    afrag[0].x = p.x * pl0; afrag[0].y = p.x * pl1;
    afrag[1].x = p.y * pl0; afrag[1].y = p.y * pl1;
    afrag[2].x = p.z * pl0; afrag[2].y = p.z * pl1;
    afrag[3].x = pl0;       afrag[3].y = pl1;

    float r[8];
    #pragma unroll
    for (int j = 0; j < 8; ++j) r[j] = 0.0f;

    #pragma unroll
    for (int t = 0; t < 4; ++t) {
      v8f c = {};
      #pragma unroll
      for (int q = 0; q < 4; ++q) {
        c = __builtin_amdgcn_wmma_f32_16x16x4_f32(
            /*neg_a=*/false, afrag[q], /*neg_b=*/false, bfrag[t][q],
            /*c_mod=*/(short)0, c, /*reuse_a=*/false, /*reuse_b=*/false);
      }
      // D: VGPR j, lane<16 -> (point tile*16+j, k=t*16+lane); lane>=16 -> point +8
      #pragma unroll
      for (int j = 0; j < 8; ++j)
        r[j] += cst[t] * __builtin_amdgcn_exp2f(c[j]);
    }

    // Butterfly transpose-reduce: 8 rows x 16 lanes (per half) -> lane l holds
    // the k-total of point (l&7) + (hi ? 8 : 0).
    #pragma unroll
    for (int j = 0; j < 4; ++j) {
      float t0 = r[2 * j]     + __shfl_xor(r[2 * j],     1);
      float t1 = r[2 * j + 1] + __shfl_xor(r[2 * j + 1], 1);
      r[j] = (lane & 1) ? t1 : t0;
    }
    #pragma unroll
    for (int j = 0; j < 2; ++j) {
      float t0 = r[2 * j]     + __shfl_xor(r[2 * j],     2);
      float t1 = r[2 * j + 1] + __shfl_xor(r[2 * j + 1], 2);
      r[j] = (lane & 2) ? t1 : t0;
    }
    {
      float t0 = r[0] + __shfl_xor(r[0], 4);
      float t1 = r[1] + __shfl_xor(r[1], 4);
      r[0] = (lane & 4) ? t1 : t0;
    }
    float sp = r[0] + __shfl_xor(r[0], 8);

    // Lane's own loaded point is (l&15); held total is for (l&7)+(hi?8:0).
    // They coincide exactly on lanes 0-7 (points 0-7) and 24-31 (points 8-15).
    float cg = p.w > 0.0f ? 1.0f : 0.0f;
    float x = SOFT_SCALE * (sp - LEVEL_SET);
    float pred = __builtin_amdgcn_rcpf(1.0f + __builtin_amdgcn_exp2f(-x * LOG2E));
    float d = cg - pred;
    float w = cg > 0.5f ? 1.0f : 10.0f;
    float contrib = w * d * d;
    bool active = (lane < 8) || (lane >= 24);
    loss_acc += active ? contrib : 0.0f;
  }

  loss_acc += __shfl_xor(loss_acc, 16);
  loss_acc += __shfl_xor(loss_acc, 8);
  loss_acc += __shfl_xor(loss_acc, 4);
  loss_acc += __shfl_xor(loss_acc, 2);
  loss_acc += __shfl_xor(loss_acc, 1);
  if (lane == 0) partials[b * wavesPerBatch + waveId] = loss_acc;
}

// ---------------------------------------------------------------------------
// Kernel 3: deterministic tree reduction of partials -> mean loss
// ---------------------------------------------------------------------------
__global__ void reduce_kernel(const float* __restrict__ part, int n,
                              float* __restrict__ out) {
  __shared__ float sm[256];
  float s = 0.0f;
  for (int i = threadIdx.x; i < n; i += 256) s += part[i];
  sm[threadIdx.x] = s;
  __syncthreads();
  for (int off = 128; off > 0; off >>= 1) {
    if ((int)threadIdx.x < off) sm[threadIdx.x] += sm[threadIdx.x + off];
    __syncthreads();
  }
  if (threadIdx.x == 0) out[0] = sm[0] * (1.0f / (float)(BB * NN));
}

extern "C" void kernel_launch(void* const* d_in, const int* in_sizes, int n_in,
                              void* d_out, int out_size, void* d_ws, size_t ws_size,
                              hipStream_t stream) {
  const float* points    = (const float*)d_in[0]; // B*N*4
  const float* constants = (const float*)d_in[1]; // B*K
  const float* scales    = (const float*)d_in[2]; // B*K*3
  const float* rotations = (const float*)d_in[3]; // B*K*3
  const float* centers   = (const float*)d_in[4]; // B*K*3
  float* out = (float*)d_out;

  // workspace layout: [coef: B*16*K floats][partials: B*wavesPerBatch floats]
  float* coef = (float*)d_ws;
  const int blocksPerBatch = 32;                   // grid.y
  const int wavesPerBatch = blocksPerBatch * 8;    // 256 waves/batch, 8 tiles each
  float* partials = coef + BB * 16 * KK;
  const int nPart = BB * wavesPerBatch;

  setup_coef_kernel<<<(BB * KK + 255) / 256, 256, 0, stream>>>(
      scales, rotations, centers, coef);

  rbf_loss_kernel<<<dim3(BB, blocksPerBatch), 256, 0, stream>>>(
      points, constants, coef, partials);

  reduce_kernel<<<1, 256, 0, stream>>>(partials, nPart, out);
}